// CrossCoderDecoder_86294482912183
// MI455X (gfx1250) — compile-verified
//
#include <hip/hip_runtime.h>

typedef __attribute__((ext_vector_type(2))) float v2f;
typedef __attribute__((ext_vector_type(8))) float v8f;

// Problem constants (match the reference).
#define BDIM 2048
#define LDIM 2
#define FDIM 65536
#define DDIM 768

#define KC    64          // K-chunk per LDS buffer (16 WMMA steps of K=4)
#define NCH   (FDIM / KC) // 1024 chunks (even)
#define WAVES 8           // 256 threads = 8 wave32
#define BN    (WAVES*16)  // 128 output columns per workgroup
#define PITCH 68          // LDS row pitch in floats: 272B (16B-aligned, conflict-free b64 reads)

__device__ __forceinline__ unsigned lds_off(const void* p) {
    // Generic LDS address: low 32 bits are the LDS byte offset (ISA 10.2 aperture rules).
    return (unsigned)(unsigned long long)p;
}

__device__ __forceinline__ void async_tile(unsigned lds, const float* g) {
    // CDNA5 async memory->LDS copy, tracked by ASYNCcnt (no VGPR round-trip).
    asm volatile("global_load_async_to_lds_b128 %0, %1, off"
                 :: "v"(lds), "v"((unsigned long long)g) : "memory");
}
__device__ __forceinline__ void wait_async_le1() {
    asm volatile("s_wait_asynccnt 0x1" ::: "memory");
}
__device__ __forceinline__ void wait_async_le0() {
    asm volatile("s_wait_asynccnt 0x0" ::: "memory");
}

// One K-chunk: 16 x (A: ds_load_b64, B: 2 coalesced global b32, WMMA f32 16x16x4).
__device__ __forceinline__ void compute_chunk(const float* ap, const float*& wp,
                                              int half, v8f& acc) {
#pragma unroll
    for (int kk = 0; kk < KC / 4; ++kk) {
        v2f a = *reinterpret_cast<const v2f*>(ap + 4 * kk + 2 * half);
        v2f b;
        b.x = wp[(size_t)(4 * kk) * DDIM];
        b.y = wp[(size_t)(4 * kk + 1) * DDIM];
        acc = __builtin_amdgcn_wmma_f32_16x16x4_f32(
            false, a, false, b, (short)0, acc, false, false);
    }
    wp += (size_t)KC * DDIM;
}

__global__ __launch_bounds__(256, 2)
void crosscoder_decoder_gemm(const float* __restrict__ f,    // [B, F]
                             const float* __restrict__ W,    // [L, F, D]
                             const float* __restrict__ bias, // [L, D]
                             float* __restrict__ out)        // [B, L, D]
{
    __shared__ __align__(16) float sA[2][16][PITCH];   // double-buffered 16 x KC tile

    const int tid  = threadIdx.x;
    const int lane = tid & 31;
    const int nl   = lane & 15;        // A row / output column within tile
    const int half = lane >> 4;        // 0: K pair {0,1}; 1: K pair {2,3}
    const int wave = tid >> 5;

    const int m0 = blockIdx.x * 16;                 // 16 output rows (batch)
    const int n0 = blockIdx.y * BN + wave * 16;     // 16 output cols (d) per wave
    const int l  = blockIdx.z;

    const float* __restrict__ Wl = W + (size_t)l * FDIM * DDIM;
    const int n = n0 + nl;

    // Per-lane async-copy slice: one b128 (4 floats) per lane per chunk.
    const int ar = tid >> 4;           // row 0..15
    const int ac = (tid & 15) << 2;    // col 0,4,...,60
    const unsigned ldsa0 = lds_off(&sA[0][ar][ac]);
    const unsigned ldsa1 = lds_off(&sA[1][ar][ac]);
    const float* gsrc = f + (size_t)(m0 + ar) * FDIM + ac;

    // Per-lane W stream pointer (k offsets fold to 24-bit immediates).
    const float* wp = Wl + (size_t)(2 * half) * DDIM + n;

    const float* ap0 = &sA[0][nl][0];
    const float* ap1 = &sA[1][nl][0];

    v8f acc = {};

    // Prologue: async-copy chunk 0 into buffer 0.
    async_tile(ldsa0, gsrc);
    gsrc += KC;

    // Steady state: branch-free, buffer index is compile-time constant.
    for (int c = 0; c < NCH - 2; c += 2) {
        async_tile(ldsa1, gsrc);        // chunk c+1 -> buf1
        gsrc += KC;
        wait_async_le1();               // chunk c resident
        __syncthreads();
        compute_chunk(ap0, wp, half, acc);
        __syncthreads();                // buf0 free for rewrite

        async_tile(ldsa0, gsrc);        // chunk c+2 -> buf0
        gsrc += KC;
        wait_async_le1();               // chunk c+1 resident
        __syncthreads();
        compute_chunk(ap1, wp, half, acc);
        __syncthreads();                // buf1 free for rewrite
    }

    // Epilogue: chunks NCH-2 (buf0) and NCH-1 (buf1).
    async_tile(ldsa1, gsrc);            // final chunk -> buf1
    wait_async_le1();
    __syncthreads();
    compute_chunk(ap0, wp, half, acc);
    __syncthreads();

    wait_async_le0();
    __syncthreads();
    compute_chunk(ap1, wp, half, acc);

    // Epilogue: add bias[l, n] (uniform over the 8 rows each lane holds).
    const float bv = bias[(size_t)l * DDIM + n];
#pragma unroll
    for (int r = 0; r < 8; ++r) {
        const int m = m0 + r + 8 * half;   // C layout: vgpr r -> row r (+8 for upper half)
        out[(size_t)m * (LDIM * DDIM) + (size_t)l * DDIM + n] = acc[r] + bv;
    }
}

extern "C" void kernel_launch(void* const* d_in, const int* in_sizes, int n_in,
                              void* d_out, int out_size, void* d_ws, size_t ws_size,
                              hipStream_t stream) {
    const float* f    = (const float*)d_in[0];   // [2048, 65536]
    const float* W    = (const float*)d_in[1];   // [2, 65536, 768]
    const float* bias = (const float*)d_in[2];   // [2, 768]
    float*       out  = (float*)d_out;           // [2048, 2, 768]

    dim3 grid(BDIM / 16, DDIM / BN, LDIM);       // (128, 6, 2)
    dim3 block(32 * WAVES);                      // 256 threads = 8 waves (wave32)

    crosscoder_decoder_gemm<<<grid, block, 0, stream>>>(f, W, bias, out);
}